// GaussLoss_6262062318269
// MI455X (gfx1250) — compile-verified
//
#include <hip/hip_runtime.h>
#include <math.h>

typedef float v2f __attribute__((ext_vector_type(2)));
typedef float v8f __attribute__((ext_vector_type(8)));

#define LOG_2PI 1.8378770664093453f

__device__ __forceinline__ float softplus_f(float x) {
    // numerically stable softplus, matches jax.nn.softplus
    return fmaxf(x, 0.0f) + log1pf(expf(-fabsf(x)));
}

// Reduce 64 values (2 per lane of a full wave32) to a single sum using
// V_WMMA_F32_16X16X4_F32 with an all-ones B matrix.
//   A (16x4 f32, 2 VGPRs): lane<16 -> (M=lane,K=0/1), lane>=16 -> (M=lane-16,K=2/3)
//   D[m][n] = rowsum(m) for every n; summing the 8 D VGPRs on lane0 gives
//   rows 0..7, on lane16 rows 8..15; shfl_xor(16) combines the halves.
// Must be executed by a fully-active wave (EXEC all ones).
__device__ __forceinline__ float wmma_reduce64(float a0, float a1) {
    v2f a;    a[0] = a0;   a[1] = a1;
    v2f ones; ones[0] = 1.0f; ones[1] = 1.0f;
    v8f c = {};
    v8f d = __builtin_amdgcn_wmma_f32_16x16x4_f32(
        /*neg_a=*/false, a, /*neg_b=*/false, ones,
        /*c_mod=*/(short)0, c, /*reuse_a=*/false, /*reuse_b=*/false);
    float r = ((d[0] + d[1]) + (d[2] + d[3])) + ((d[4] + d[5]) + (d[6] + d[7]));
    r += __shfl_xor(r, 16, 32);
    return r; // full sum, broadcast to all lanes
}

// Pass 1: each block streams a 16-row x 1024-col tile of pred/gt, accumulates
// sum((gt-pred)^2 * inv_var) and writes one partial per block (deterministic).
__global__ __launch_bounds__(256)
void quad_partial_kernel(const float* __restrict__ pred,
                         const float* __restrict__ gt,
                         const float* __restrict__ sigma,
                         float* __restrict__ partials,
                         int N, int rowsPerBlock) {
    const int tid     = threadIdx.x;
    const int colBase = blockIdx.x * (blockDim.x * 4) + tid * 4;
    const int row0    = blockIdx.y * rowsPerBlock;

    // Per-column inverse variance, hoisted out of the row loop.
    float4 s = *reinterpret_cast<const float4*>(sigma + colBase);
    const float iv0 = 1.0f / softplus_f(s.x);
    const float iv1 = 1.0f / softplus_f(s.y);
    const float iv2 = 1.0f / softplus_f(s.z);
    const float iv3 = 1.0f / softplus_f(s.w);

    const size_t base    = ((size_t)row0 * (size_t)N + (size_t)colBase) >> 2;
    const size_t stride4 = (size_t)N >> 2; // float4s per row
    const float4* __restrict__ p4 = reinterpret_cast<const float4*>(pred) + base;
    const float4* __restrict__ g4 = reinterpret_cast<const float4*>(gt)   + base;

    float acc0 = 0.0f, acc1 = 0.0f, acc2 = 0.0f, acc3 = 0.0f;
#pragma unroll 4
    for (int r = 0; r < rowsPerBlock; ++r) {
        float4 pv = p4[(size_t)r * stride4];
        float4 gv = g4[(size_t)r * stride4];
        float d0 = gv.x - pv.x;
        float d1 = gv.y - pv.y;
        float d2 = gv.z - pv.z;
        float d3 = gv.w - pv.w;
        acc0 = fmaf(d0 * d0, iv0, acc0);
        acc1 = fmaf(d1 * d1, iv1, acc1);
        acc2 = fmaf(d2 * d2, iv2, acc2);
        acc3 = fmaf(d3 * d3, iv3, acc3);
    }
    float acc = (acc0 + acc1) + (acc2 + acc3);

    __shared__ float red[256];
    red[tid] = acc;
    __syncthreads();

    if (tid < 32) { // wave 0, fully active (wave32) -> EXEC all ones for WMMA
        const float* p = red + tid * 8;
        float a0 = (p[0] + p[1]) + (p[2] + p[3]);
        float a1 = (p[4] + p[5]) + (p[6] + p[7]);
        float tot = wmma_reduce64(a0, a1);
        if (tid == 0)
            partials[blockIdx.y * gridDim.x + blockIdx.x] = tot;
    }
}

// Pass 2 (single block): deterministic final sum of block partials, plus the
// log-det and constant terms. Same WMMA reduction for the last 256 -> 1.
__global__ __launch_bounds__(256)
void finalize_kernel(const float* __restrict__ partials, int nPartials,
                     const float* __restrict__ sigma, int N, int Brows,
                     float* __restrict__ out) {
    const int tid = threadIdx.x;

    float q = 0.0f;
    for (int i = tid; i < nPartials; i += 256)
        q += partials[i];

    float l = 0.0f;
    for (int i = tid; i < N; i += 256)
        l += logf(softplus_f(sigma[i]));

    float s = 0.5f * q + 0.5f * (float)Brows * l;
    if (tid == 0)
        s += 0.5f * (float)Brows * (float)N * LOG_2PI;

    __shared__ float red[256];
    red[tid] = s;
    __syncthreads();

    if (tid < 32) {
        const float* p = red + tid * 8;
        float a0 = (p[0] + p[1]) + (p[2] + p[3]);
        float a1 = (p[4] + p[5]) + (p[6] + p[7]);
        float tot = wmma_reduce64(a0, a1);
        if (tid == 0)
            out[0] = tot;
    }
}

extern "C" void kernel_launch(void* const* d_in, const int* in_sizes, int n_in,
                              void* d_out, int out_size, void* d_ws, size_t ws_size,
                              hipStream_t stream) {
    const float* pred  = (const float*)d_in[0];
    const float* gt    = (const float*)d_in[1];
    const float* sigma = (const float*)d_in[2];
    float* out = (float*)d_out;
    float* partials = (float*)d_ws;

    const int N     = in_sizes[2];             // 4096
    const int Brows = in_sizes[0] / N;         // 8192

    const int threads       = 256;
    const int colsPerBlock  = threads * 4;     // 1024
    const int rowsPerBlock  = 16;
    const int gx = N / colsPerBlock;           // 4
    const int gy = Brows / rowsPerBlock;       // 512  -> 2048 blocks total

    dim3 grid(gx, gy);
    quad_partial_kernel<<<grid, threads, 0, stream>>>(pred, gt, sigma, partials,
                                                      N, rowsPerBlock);
    finalize_kernel<<<1, threads, 0, stream>>>(partials, gx * gy, sigma, N, Brows, out);
}